// AleatoricUncertaintyEstimator_63651415327368
// MI455X (gfx1250) — compile-verified
//
#include <hip/hip_runtime.h>
#include <hip/hip_bf16.h>

typedef __attribute__((ext_vector_type(4))) float v4f;
typedef __attribute__((ext_vector_type(2))) float v2f;
typedef __attribute__((ext_vector_type(8))) float v8f;

#define NB 8192
#define KTOP 10
#define STRIPES 16
#define ROWS_PER_STRIPE (NB / STRIPES) /* 512 */
#define NEG_BIG (-1.0e30f)
#define INV_T 50.0f              /* 1 / 0.02 */
#define LOG_MAX_ENT 9.0109133f   /* ln(8192 + 1e-10) */

__device__ __forceinline__ void topk_insert(float (&tv)[KTOP], int (&ti)[KTOP],
                                            float x, int c) {
  if (x > tv[KTOP - 1]) {
    tv[KTOP - 1] = x; ti[KTOP - 1] = c;
#pragma unroll
    for (int k = KTOP - 1; k > 0; --k) {
      if (tv[k] > tv[k - 1]) {
        float tf = tv[k]; tv[k] = tv[k - 1]; tv[k - 1] = tf;
        int tt = ti[k]; ti[k] = ti[k - 1]; ti[k - 1] = tt;
      }
    }
  }
}

// ---------------- Kernel 1: per-row online-softmax entropy + row top-10 -----
// One wave per row; single streaming pass (non-temporal) over the row.
__global__ __launch_bounds__(256) void row_pass(const float* __restrict__ sim,
                                                float* __restrict__ ne_out,
                                                int* __restrict__ t2i_idx) {
  const int lane = threadIdx.x & 31;
  const int wave = threadIdx.x >> 5;
  const int row = blockIdx.x * 8 + wave;
  const v4f* rp = (const v4f*)(sim + (size_t)row * NB);

  float m = NEG_BIG, Z = 0.0f, W = 0.0f;
  float tv[KTOP]; int ti[KTOP];
#pragma unroll
  for (int k = 0; k < KTOP; ++k) { tv[k] = NEG_BIG; ti[k] = 0; }

  auto push = [&](float x, int c) {
    topk_insert(tv, ti, x, c);
    float s = x * INV_T;
    if (s > m) {                        // new running max: rescale Z, W
      float r = __expf(m - s);
      W = r * (W + (m - s) * Z);
      Z = r * Z + 1.0f;
      m = s;
    } else if (s > m - 40.0f) {         // below this, e < 4e-18: no effect on Z>=1
      float e = __expf(s - m);
      Z += e;
      W += e * (s - m);
    }
  };

  for (int i = lane; i < NB / 4; i += 64) {   // 2x unrolled float4 stream
    v4f a = __builtin_nontemporal_load(rp + i);
    v4f b = __builtin_nontemporal_load(rp + i + 32);
    int c0 = i * 4, c1 = (i + 32) * 4;
    push(a.x, c0);     push(a.y, c0 + 1); push(a.z, c0 + 2); push(a.w, c0 + 3);
    push(b.x, c1);     push(b.y, c1 + 1); push(b.z, c1 + 2); push(b.w, c1 + 3);
  }

  // Butterfly merge of online-softmax state (exact associative combine).
#pragma unroll
  for (int off = 16; off >= 1; off >>= 1) {
    float om = __shfl_xor(m, off, 32);
    float oZ = __shfl_xor(Z, off, 32);
    float oW = __shfl_xor(W, off, 32);
    float mm = fmaxf(m, om);
    float r1 = __expf(m - mm), r2 = __expf(om - mm);
    W = r1 * (W + (m - mm) * Z) + r2 * (oW + (om - mm) * oZ);
    Z = r1 * Z + r2 * oZ;
    m = mm;
  }
  if (lane == 0) {
    float H = __logf(Z) - W / Z;        // H = -sum p*ln(p)
    ne_out[row] = H / LOG_MAX_ENT;
  }

  // Merge 32 sorted per-lane top-10 lists: 10 rounds of argmax + pop.
  for (int sel = 0; sel < KTOP; ++sel) {
    float hv = tv[0]; int hl = lane;
#pragma unroll
    for (int off = 16; off >= 1; off >>= 1) {
      float ov = __shfl_xor(hv, off, 32);
      int   ol = __shfl_xor(hl, off, 32);
      if (ov > hv || (ov == hv && ol < hl)) { hv = ov; hl = ol; }
    }
    int hcol = __shfl(ti[0], hl, 32);
    if (lane == 0) t2i_idx[row * KTOP + sel] = hcol;
    if (lane == hl) {                   // winner pops its head
#pragma unroll
      for (int k = 0; k < KTOP - 1; ++k) { tv[k] = tv[k + 1]; ti[k] = ti[k + 1]; }
      tv[KTOP - 1] = NEG_BIG;
    }
  }
}

// ---------------- Kernel 2: per-column top-10, row-striped for MLP ----------
__global__ __launch_bounds__(256) void col_pass(const float* __restrict__ sim,
                                                float* __restrict__ cp_val,
                                                int* __restrict__ cp_idx) {
  const int col = blockIdx.x * 256 + threadIdx.x;
  const int r0 = blockIdx.y * ROWS_PER_STRIPE;
  float tv[KTOP]; int ti[KTOP];
#pragma unroll
  for (int k = 0; k < KTOP; ++k) { tv[k] = NEG_BIG; ti[k] = 0; }

  const float* p = sim + (size_t)r0 * NB + col;
  for (int r = 0; r < ROWS_PER_STRIPE; r += 8) {
    if (r + 24 < ROWS_PER_STRIPE)
      __builtin_prefetch(p + (size_t)(r + 24) * NB, 0, 0);  // global_prefetch_b8
    float x[8];
#pragma unroll
    for (int j = 0; j < 8; ++j)
      x[j] = __builtin_nontemporal_load(p + (size_t)(r + j) * NB);
#pragma unroll
    for (int j = 0; j < 8; ++j)
      topk_insert(tv, ti, x[j], r0 + r + j);
  }
  size_t o = ((size_t)blockIdx.y * NB + col) * KTOP;
#pragma unroll
  for (int k = 0; k < KTOP; ++k) { cp_val[o + k] = tv[k]; cp_idx[o + k] = ti[k]; }
}

// ---------------- Kernel 2b: merge stripe candidates -> exact column top-10 -
__global__ __launch_bounds__(256) void col_merge(const float* __restrict__ cp_val,
                                                 const int* __restrict__ cp_idx,
                                                 int* __restrict__ i2t_idx) {
  const int col = blockIdx.x * 256 + threadIdx.x;
  float v[STRIPES * KTOP]; int id[STRIPES * KTOP];
  for (int s = 0; s < STRIPES; ++s) {
    size_t o = ((size_t)s * NB + col) * KTOP;
#pragma unroll
    for (int k = 0; k < KTOP; ++k) {
      v[s * KTOP + k] = cp_val[o + k];
      id[s * KTOP + k] = cp_idx[o + k];
    }
  }
  for (int sel = 0; sel < KTOP; ++sel) {
    float best = -__builtin_inff(); int bi = 0, bp = 0;
    for (int e = 0; e < STRIPES * KTOP; ++e)
      if (v[e] > best) { best = v[e]; bi = id[e]; bp = e; }
    v[bp] = -__builtin_inff();
    i2t_idx[col * KTOP + sel] = bi;   // ascending-row tie order matches top_k
  }
}

// ---------------- Kernel 3: overlap + WMMA affine blend ---------------------
// u_i = -0.5*ra_i + 0.5*ne_i + 0.5 for 16 rows per wave via
// V_WMMA_F32_16X16X4_F32: A = [ra ne 1 0] rows, B = weight column (bcast).
__global__ __launch_bounds__(256) void combine(const float* __restrict__ ne_arr,
                                               const int* __restrict__ t2i,
                                               const int* __restrict__ i2t,
                                               float* __restrict__ out) {
  const int lane = threadIdx.x & 31;
  const int wave = threadIdx.x >> 5;
  const int base = blockIdx.x * 128 + wave * 16;
  const int row = base + (lane & 15);

  int a[KTOP], b[KTOP];
#pragma unroll
  for (int k = 0; k < KTOP; ++k) { a[k] = t2i[row * KTOP + k]; b[k] = i2t[row * KTOP + k]; }
  int ov = 0;
#pragma unroll
  for (int i = 0; i < KTOP; ++i)
#pragma unroll
    for (int j = 0; j < KTOP; ++j)
      ov += (a[i] == b[j]) ? 1 : 0;
  float ra = (float)ov * (1.0f / KTOP);
  float ne = ne_arr[row];

  const bool lo = lane < 16;
  // A 16x4 f32: lane l%16 = row M; lane halves split the K pairs.
  v2f A;  A.x  = lo ? ra    : 1.0f;  A.y  = lo ? ne   : 0.0f;
  // B 4x16 f32: same lane-half -> K-set mapping, so products pair correctly.
  v2f Bm; Bm.x = lo ? -0.5f : 0.5f;  Bm.y = lo ? 0.5f : 0.0f;
  v8f C = {0.f, 0.f, 0.f, 0.f, 0.f, 0.f, 0.f, 0.f};
  v8f D = __builtin_amdgcn_wmma_f32_16x16x4_f32(
      /*neg_a=*/false, A, /*neg_b=*/false, Bm,
      /*c_mod=*/(short)0, C, /*reuse_a=*/false, /*reuse_b=*/false);

  // C/D layout: VGPR r, lane l -> M = r + 8*(l>=16), N = l%16. All N equal.
  if (lane == 0) {
#pragma unroll
    for (int j = 0; j < 8; ++j) out[base + j] = D[j];
  } else if (lane == 16) {
#pragma unroll
    for (int j = 0; j < 8; ++j) out[base + 8 + j] = D[j];
  }
  if (lo) out[NB + row] = ne;
}

extern "C" void kernel_launch(void* const* d_in, const int* in_sizes, int n_in,
                              void* d_out, int out_size, void* d_ws, size_t ws_size,
                              hipStream_t stream) {
  (void)in_sizes; (void)n_in; (void)out_size; (void)ws_size;
  const float* sim = (const float*)d_in[0];   // d_in[1] (pids) unused by reference math
  float* out = (float*)d_out;

  // Workspace layout (~11.2 MB total)
  float* ws_ne  = (float*)d_ws;                          // NB floats
  int*   ws_t2i = (int*)(ws_ne + NB);                    // NB*KTOP ints
  int*   ws_i2t = ws_t2i + (size_t)NB * KTOP;            // NB*KTOP ints
  float* ws_cpv = (float*)(ws_i2t + (size_t)NB * KTOP);  // STRIPES*NB*KTOP floats
  int*   ws_cpi = (int*)(ws_cpv + (size_t)STRIPES * NB * KTOP);

  row_pass<<<NB / 8, 256, 0, stream>>>(sim, ws_ne, ws_t2i);
  col_pass<<<dim3(NB / 256, STRIPES), 256, 0, stream>>>(sim, ws_cpv, ws_cpi);
  col_merge<<<NB / 256, 256, 0, stream>>>(ws_cpv, ws_cpi, ws_i2t);
  combine<<<NB / 128, 256, 0, stream>>>(ws_ne, ws_t2i, ws_i2t, out);
}